// MyLayer_45930380264256
// MI455X (gfx1250) — compile-verified
//
#include <hip/hip_runtime.h>

// out[row, 0:64] = x[row, 128:192] / max(||x[row,128:192]||_2, 1e-12)
// x: (batch, 192) f32, out: (batch, 64) f32. A/b/B/x0 inputs are dead code
// in the reference (u depends only on x).

typedef __attribute__((ext_vector_type(2))) float v2f;
typedef __attribute__((ext_vector_type(4))) float v4f;
typedef __attribute__((ext_vector_type(8))) float v8f;

namespace {
constexpr int kN      = 64;   // vector length per step
constexpr int kStride = 192;  // NUM_STEPS * n
constexpr int kOff    = 128;  // offset of last segment
constexpr float kEps  = 1e-12f;
}

__global__ __launch_bounds__(256, 8)
void ellipsoid_norm_kernel(const float* __restrict__ x,
                           float* __restrict__ out,
                           int batch)
{
    const int lane    = threadIdx.x & 31;
    const int wave    = (blockIdx.x * blockDim.x + threadIdx.x) >> 5;
    const int rowBase = wave * 16;               // 16 rows per wave
    if (rowBase >= batch) return;                // wave-uniform: EXEC stays all-ones

    const int hi = lane >> 4;                    // lane half
    const int m  = lane & 15;                    // row within tile (A/C lane->M map)

    // ---------- Phase 1: row norms^2 via V_WMMA_F32_16X16X4_F32 ----------
    // A holds x^2 for 16 rows (16x4 f32 tile, K advanced by 4 per step).
    // B = all-ones  =>  D[M][N] = sum_k x[M][k]^2 in EVERY column N.
    // The sum is invariant to any K-permutation inside the A layout, so the
    // only layout facts relied on are the documented lane->M mappings.
    int arow = rowBase + m;
    if (arow >= batch) arow = batch - 1;         // tail clamp (full EXEC for WMMA)
    const float* rp = x + (size_t)arow * kStride + kOff + hi * 2;

    v2f ones; ones.x = 1.0f; ones.y = 1.0f;
    v8f acc = {};
#pragma unroll
    for (int t = 0; t < 16; ++t) {
        v2f v = *(const v2f*)(rp + 4 * t);       // 8B load, 8B aligned
        v2f a; a.x = v.x * v.x; a.y = v.y * v.y;
        acc = __builtin_amdgcn_wmma_f32_16x16x4_f32(
            /*neg_a=*/false, a, /*neg_b=*/false, ones,
            /*c_mod=*/(short)0, acc, /*reuse_a=*/false, /*reuse_b=*/false);
    }

    // C/D layout: lane l, component c holds D[c + 8*hi][l&15]; all columns of a
    // row are identical, so this lane half owns norms^2 of rows (8*hi .. 8*hi+7).
    float inv[8];
#pragma unroll
    for (int c = 0; c < 8; ++c)
        inv[c] = 1.0f / fmaxf(sqrtf(acc[c]), kEps);

    // ---------- Phase 2: scale + store, 512B contiguous per wave-step ----------
    // Step j: lanes 0-15 cover row rowBase+2j (cols m*4..m*4+3),
    //         lanes 16-31 cover row rowBase+2j+1 -> out writes are a single
    //         contiguous 512B burst per step; input reads are two 256B bursts.
#pragma unroll
    for (int j = 0; j < 8; ++j) {
        const int src = (j < 4) ? 0 : 16;        // lane half owning rows 2j, 2j+1
        float invA = __shfl(inv[(2 * j) & 7],     src, 32);  // row 2j
        float invB = __shfl(inv[(2 * j + 1) & 7], src, 32);  // row 2j+1
        float s = hi ? invB : invA;
        int r = rowBase + 2 * j + hi;
        if (r < batch) {
            v4f v = *(const v4f*)(x + (size_t)r * kStride + kOff + m * 4);
            v.x *= s; v.y *= s; v.z *= s; v.w *= s;
            *(v4f*)(out + (size_t)r * kN + m * 4) = v;
        }
    }
}

extern "C" void kernel_launch(void* const* d_in, const int* in_sizes, int n_in,
                              void* d_out, int out_size, void* d_ws, size_t ws_size,
                              hipStream_t stream)
{
    const float* x = (const float*)d_in[0];      // inputs: x, A, b, B, x0 (only x used)
    float* out     = (float*)d_out;

    const int batch  = in_sizes[0] / kStride;    // 8192 for the reference setup
    const int waves  = (batch + 15) / 16;        // 16 rows per wave
    const int blocks = (waves + 7) / 8;          // 8 waves (256 threads) per block

    ellipsoid_norm_kernel<<<blocks, 256, 0, stream>>>(x, out, batch);
}